// RNN_single_gate_90958817394761
// MI455X (gfx1250) — compile-verified
//
#include <hip/hip_runtime.h>

// Problem constants (match reference)
#define B_    64
#define S_    512
#define E_    1024
#define H_    1024
#define OUT_  32000
#define CHUNK_ 64          // timesteps per pipeline chunk
#define RNN_WGS 64         // persistent workgroups in recurrence (16 cols each)

// Workspace layout (requires ~17 MB):
//   XW chunk buffer : CHUNK_*B_*H_ floats (16 MB)
//   h double buffer : 2*B_*H_ floats (512 KB)
//   barrier counter : 1 int

typedef __attribute__((ext_vector_type(2))) float v2f;
typedef __attribute__((ext_vector_type(8))) float v8f;

__device__ __forceinline__ v8f wmma_f32_16x16x4(v2f a, v2f b, v8f c) {
  // 8 args: (neg_a, A, neg_b, B, c_mod, C, reuse_a, reuse_b)
  return __builtin_amdgcn_wmma_f32_16x16x4_f32(false, a, false, b, (short)0, c,
                                               false, false);
}

// ---------------------------------------------------------------------------
// Phase 1: XW[t,b,:] = emb[words[b,t]] @ wx + bias   (one 64-step chunk)
// grid = 64 (timesteps) x 4 (256-col blocks); 256 thr = 8 waves.
// Each wave: 64x32 output block = 4 row-tiles x 2 col-tiles (8 WMMA / 6 loads).
// ---------------------------------------------------------------------------
__global__ __launch_bounds__(256)
void xw_gemm_kernel(const int* __restrict__ words, const float* __restrict__ emb,
                    const float* __restrict__ wx, const float* __restrict__ bias,
                    float* __restrict__ XW, int t_base) {
  __shared__ int widx[B_];
  const int tid  = threadIdx.x;
  const int lane = tid & 31;
  const int wave = tid >> 5;              // 0..7
  const int rb   = (int)blockIdx.x >> 2;  // timestep within chunk (0..63)
  const int cbl  = (int)blockIdx.x & 3;   // 256-col block (0..3)
  const int t    = t_base + rb;

  if (tid < B_) widx[tid] = words[tid * S_ + t];   // words is [B, S]
  __syncthreads();

  const int frag_mn = lane & 15;          // A: M index / B: N index
  const int frag_k  = (lane >> 4) << 1;   // K base (0 or 2)
  const int ncol0   = cbl * 256 + wave * 32;       // wave's first column

  const float* arow[4];
#pragma unroll
  for (int rt = 0; rt < 4; ++rt)
    arow[rt] = emb + (size_t)widx[rt * 16 + frag_mn] * E_;
  const float* bcol0 = wx + ncol0 + frag_mn;
  const float* bcol1 = wx + ncol0 + 16 + frag_mn;

  v8f acc[4][2] = {};
  for (int k0 = 0; k0 < E_; k0 += 4) {
    v2f a[4];
#pragma unroll
    for (int rt = 0; rt < 4; ++rt)
      a[rt] = *(const v2f*)(arow[rt] + k0 + frag_k);
    v2f b0, b1;
    b0.x = bcol0[(size_t)(k0 + frag_k) * H_];
    b0.y = bcol0[(size_t)(k0 + frag_k + 1) * H_];
    b1.x = bcol1[(size_t)(k0 + frag_k) * H_];
    b1.y = bcol1[(size_t)(k0 + frag_k + 1) * H_];
#pragma unroll
    for (int rt = 0; rt < 4; ++rt) {
      acc[rt][0] = wmma_f32_16x16x4(a[rt], b0, acc[rt][0]);
      acc[rt][1] = wmma_f32_16x16x4(a[rt], b1, acc[rt][1]);
    }
  }

  const float bv0 = bias[ncol0 + frag_mn];
  const float bv1 = bias[ncol0 + 16 + frag_mn];
  float* out0 = XW + (size_t)(rb * B_) * H_ + ncol0 + frag_mn;
#pragma unroll
  for (int rt = 0; rt < 4; ++rt) {
    const int mbase = rt * 16 + (lane >> 4) * 8;    // C/D: M = v + 8*(lane/16)
#pragma unroll
    for (int v = 0; v < 8; ++v) {
      out0[(size_t)(mbase + v) * H_]      = acc[rt][0][v] + bv0;
      out0[(size_t)(mbase + v) * H_ + 16] = acc[rt][1][v] + bv1;
    }
  }
}

// ---------------------------------------------------------------------------
// Phase 2: persistent recurrence over one chunk.
// 64 WGs x 128 thr (4 waves). WG j owns cols [16j,16j+16); wh tile in LDS with
// K-pairs interleaved so B fragments are single conflict-free ds_load_b64.
// step t: h_new[:,cols] = tanh(XW[t,:,cols] + h_old @ wh[:,cols]); grid barrier.
// ---------------------------------------------------------------------------
__global__ __launch_bounds__(128)
void rnn_chunk_kernel(const float* __restrict__ XW, const float* __restrict__ wh,
                      float* __restrict__ hbuf, int* __restrict__ bar, int t_base) {
  extern __shared__ float whl[];          // [E_/2][16][2] = 64 KB
  const int tid  = threadIdx.x;
  const int lane = tid & 31;
  const int wave = tid >> 5;              // 0..3 -> 16-row tiles covering B=64
  const int cb   = (int)blockIdx.x;       // 16-col block (0..63)

  for (int idx = tid; idx < E_ * 16; idx += 128) {
    const int k = idx >> 4, c = idx & 15;
    whl[((k >> 1) * 16 + c) * 2 + (k & 1)] = wh[(size_t)k * H_ + cb * 16 + c];
  }
  __syncthreads();

  const int frag_mn = lane & 15;
  const int frag_k  = (lane >> 4) << 1;
  const int m0    = wave * 16;
  const int ncol  = cb * 16 + frag_mn;
  const int mbase = m0 + (lane >> 4) * 8;
  // B-fragment LDS base (float index) for k0=0; advances by 32 floats per k-pair
  const float* bbase = whl + (((frag_k >> 1) * 16) + frag_mn) * 2;

  for (int i = 0; i < CHUNK_; ++i) {
    const int t = t_base + i;
    const float* hin  = hbuf + (size_t)(t & 1) * (B_ * H_);
    float*       hout = hbuf + (size_t)((t + 1) & 1) * (B_ * H_);
    const float* xrow = XW + (size_t)(i * B_) * H_ + ncol;

    v8f acc0; v8f acc1 = {};
#pragma unroll
    for (int v = 0; v < 8; ++v) acc0[v] = xrow[(size_t)(mbase + v) * H_];

    const float* arow = hin + (size_t)(m0 + frag_mn) * H_;
    for (int k0 = 0; k0 < H_; k0 += 8) {
      v2f a0 = *(const v2f*)(arow + k0 + frag_k);
      v2f b0 = *(const v2f*)(bbase + k0 * 16);          // (k0+frag_k)/2 pairs
      acc0 = wmma_f32_16x16x4(a0, b0, acc0);
      v2f a1 = *(const v2f*)(arow + k0 + 4 + frag_k);
      v2f b1 = *(const v2f*)(bbase + (k0 + 4) * 16);
      acc1 = wmma_f32_16x16x4(a1, b1, acc1);
    }

    float* orow = hout + ncol;
#pragma unroll
    for (int v = 0; v < 8; ++v)
      orow[(size_t)(mbase + v) * H_] = tanhf(acc0[v] + acc1[v]);

    // ---- device-scope barrier across RNN_WGS workgroups ----
    __builtin_amdgcn_fence(__ATOMIC_RELEASE, "agent");  // flush h stores to L2
    __syncthreads();
    if (tid == 0) {
      __hip_atomic_fetch_add(bar, 1, __ATOMIC_RELAXED, __HIP_MEMORY_SCOPE_AGENT);
      const int target = (i + 1) * RNN_WGS;
      while (__hip_atomic_load(bar, __ATOMIC_RELAXED, __HIP_MEMORY_SCOPE_AGENT) < target)
        __builtin_amdgcn_s_sleep(2);
    }
    __syncthreads();
    __builtin_amdgcn_fence(__ATOMIC_ACQUIRE, "agent"); // invalidate so fresh h is read
  }
}

// ---------------------------------------------------------------------------
// Phase 3: Y = h_final @ wy + wb.  grid = OUT_/256 WGs; 8 waves.
// Each wave: 64x32 block (4 row-tiles x 2 col-tiles). wy/Y are streamed once ->
// nontemporal (TH=NT) so they do not evict L2-resident wx/wh/XW.
// ---------------------------------------------------------------------------
__global__ __launch_bounds__(256)
void y_gemm_kernel(const float* __restrict__ hfin, const float* __restrict__ wy,
                   const float* __restrict__ wb, float* __restrict__ Y) {
  const int tid  = threadIdx.x;
  const int lane = tid & 31;
  const int wave = tid >> 5;
  const int frag_mn = lane & 15;
  const int frag_k  = (lane >> 4) << 1;
  const int ncol0   = (int)blockIdx.x * 256 + wave * 32;

  const float* arow[4];
#pragma unroll
  for (int rt = 0; rt < 4; ++rt)
    arow[rt] = hfin + (size_t)(rt * 16 + frag_mn) * H_;
  const float* bcol0 = wy + ncol0 + frag_mn;
  const float* bcol1 = wy + ncol0 + 16 + frag_mn;

  v8f acc[4][2] = {};
  for (int k0 = 0; k0 < H_; k0 += 4) {
    v2f a[4];
#pragma unroll
    for (int rt = 0; rt < 4; ++rt)
      a[rt] = *(const v2f*)(arow[rt] + k0 + frag_k);
    v2f b0, b1;
    b0.x = __builtin_nontemporal_load(bcol0 + (size_t)(k0 + frag_k) * OUT_);
    b0.y = __builtin_nontemporal_load(bcol0 + (size_t)(k0 + frag_k + 1) * OUT_);
    b1.x = __builtin_nontemporal_load(bcol1 + (size_t)(k0 + frag_k) * OUT_);
    b1.y = __builtin_nontemporal_load(bcol1 + (size_t)(k0 + frag_k + 1) * OUT_);
#pragma unroll
    for (int rt = 0; rt < 4; ++rt) {
      acc[rt][0] = wmma_f32_16x16x4(a[rt], b0, acc[rt][0]);
      acc[rt][1] = wmma_f32_16x16x4(a[rt], b1, acc[rt][1]);
    }
  }

  const float bv0 = wb[ncol0 + frag_mn];
  const float bv1 = wb[ncol0 + 16 + frag_mn];
  float* out0 = Y + ncol0 + frag_mn;
#pragma unroll
  for (int rt = 0; rt < 4; ++rt) {
    const int mbase = rt * 16 + (lane >> 4) * 8;
#pragma unroll
    for (int v = 0; v < 8; ++v) {
      __builtin_nontemporal_store(acc[rt][0][v] + bv0,
                                  out0 + (size_t)(mbase + v) * OUT_);
      __builtin_nontemporal_store(acc[rt][1][v] + bv1,
                                  out0 + (size_t)(mbase + v) * OUT_ + 16);
    }
  }
}

__global__ void init_h_kernel(float* __restrict__ hbuf) {
  const int n = 2 * B_ * H_;
  for (int i = blockIdx.x * blockDim.x + threadIdx.x; i < n;
       i += gridDim.x * blockDim.x)
    hbuf[i] = 0.0f;
}

__global__ void reset_bar_kernel(int* __restrict__ bar) { *bar = 0; }

// ---------------------------------------------------------------------------
extern "C" void kernel_launch(void* const* d_in, const int* in_sizes, int n_in,
                              void* d_out, int out_size, void* d_ws, size_t ws_size,
                              hipStream_t stream) {
  (void)in_sizes; (void)n_in; (void)out_size; (void)ws_size;
  const int*   words = (const int*)d_in[0];
  const float* emb   = (const float*)d_in[1];
  const float* wx    = (const float*)d_in[2];
  const float* wh    = (const float*)d_in[3];
  const float* bias  = (const float*)d_in[4];
  const float* wy    = (const float*)d_in[5];
  const float* wb    = (const float*)d_in[6];
  float* Y = (float*)d_out;

  float* XW   = (float*)d_ws;                       // 16 MB
  float* hbuf = XW + (size_t)CHUNK_ * B_ * H_;      // 512 KB (double buffered h)
  int*   bar  = (int*)(hbuf + 2 * B_ * H_);

  init_h_kernel<<<dim3(128), dim3(256), 0, stream>>>(hbuf);

  for (int c = 0; c < S_ / CHUNK_; ++c) {
    const int t_base = c * CHUNK_;
    xw_gemm_kernel<<<dim3(64 * 4), dim3(256), 0, stream>>>(words, emb, wx, bias,
                                                           XW, t_base);
    reset_bar_kernel<<<dim3(1), dim3(1), 0, stream>>>(bar);
    rnn_chunk_kernel<<<dim3(RNN_WGS), dim3(128),
                       E_ * 16 * sizeof(float), stream>>>(XW, wh, hbuf, bar,
                                                          t_base);
  }
  // 512 steps end with final h in buffer 0
  y_gemm_kernel<<<dim3(OUT_ / 256), dim3(256), 0, stream>>>(hbuf, wy, wb, Y);
}